// EnhancedUltra_88021059764629
// MI455X (gfx1250) — compile-verified
//
#include <hip/hip_runtime.h>
#include <stdint.h>

// ---------------------------------------------------------------------------
// EnhancedUltra gate computation for MI455X (gfx1250, wave32, WMMA).
// ---------------------------------------------------------------------------

#define TBL 1024  // hash table slots (power of two, >= 4*B)

typedef __attribute__((ext_vector_type(2))) float v2f;
typedef __attribute__((ext_vector_type(8))) float v8f;

__device__ __forceinline__ uint32_t hash_node(uint32_t x) {
  x *= 2654435761u;
  x ^= x >> 15;
  return x & (TBL - 1);
}

// ---------------------------------------------------------------------------
// Kernel 1: zero Mq[B*R] and rel_counts[R] (contiguous in workspace)
// ---------------------------------------------------------------------------
__global__ void k_zero(uint32_t* p, int n) {
  int stride = gridDim.x * blockDim.x;
  for (int i = blockIdx.x * blockDim.x + threadIdx.x; i < n; i += stride)
    p[i] = 0u;
}

// ---------------------------------------------------------------------------
// Kernel 2: build node->min(batch index) hash table for the query entities.
// Duplicate query entities map to one representative row (minimum b).
// ---------------------------------------------------------------------------
__global__ void k_table(const int* __restrict__ qents, int B,
                        int* __restrict__ key, int* __restrict__ val) {
  int tid = threadIdx.x;
  for (int i = tid; i < TBL; i += blockDim.x) {
    key[i] = -1;
    val[i] = 0x7fffffff;
  }
  __syncthreads();
  if (tid < B) {
    int e = qents[tid];
    uint32_t idx = hash_node((uint32_t)e);
    for (;;) {
      int old = atomicCAS(&key[idx], -1, e);
      if (old == -1 || old == e) {
        atomicMin(&val[idx], tid);
        break;
      }
      idx = (idx + 1) & (TBL - 1);
    }
  }
}

// ---------------------------------------------------------------------------
// Kernel 3: single streaming pass over edges (the runtime dominator:
// ~38 MB of edge data @ 23.3 TB/s).
//  - 4 edges per thread via int4 (global_load_b128), prefetch lookahead
//  - LDS-privatized 512-bin relation histogram
//  - hash-probe both endpoints against the query-entity table (table in LDS)
//  - rare global float atomics into Mq rows (representative rows only)
// ---------------------------------------------------------------------------
__device__ __forceinline__ int probe_lds(const int* skey, const int* sval, int node) {
  uint32_t idx = hash_node((uint32_t)node);
  for (;;) {
    int k = skey[idx];
    if (k == node) return sval[idx];
    if (k == -1) return -1;
    idx = (idx + 1) & (TBL - 1);
  }
}

__device__ __forceinline__ void process_edge(int h, int t, int ty, int R,
                                             uint32_t* shist,
                                             const int* skey, const int* sval,
                                             float* Mq) {
  atomicAdd(&shist[ty], 1u);
  int bh = probe_lds(skey, sval, h);
  if (bh >= 0) atomicAdd(&Mq[(size_t)bh * R + ty], 1.0f);
  if (h != t) {
    int bt = probe_lds(skey, sval, t);
    if (bt >= 0) atomicAdd(&Mq[(size_t)bt * R + ty], 1.0f);
  }
}

__global__ void k_edges(const int* __restrict__ eidx, const int* __restrict__ etype,
                        int E, int R,
                        const int* __restrict__ gkey, const int* __restrict__ gval,
                        float* __restrict__ Mq, uint32_t* __restrict__ cnt) {
  __shared__ uint32_t shist[TBL];  // R <= TBL
  __shared__ int skey[TBL];
  __shared__ int sval[TBL];
  int tid = threadIdx.x, bdim = blockDim.x;
  for (int i = tid; i < R; i += bdim) shist[i] = 0u;
  for (int i = tid; i < TBL; i += bdim) {
    skey[i] = gkey[i];
    sval[i] = gval[i];
  }
  __syncthreads();

  int gsize = gridDim.x * bdim;
  if ((E & 3) == 0) {
    // Vectorized path: 4 edges per thread, 16-byte aligned b128 loads.
    int nquad = E >> 2;
    const int4* h4 = (const int4*)eidx;
    const int4* t4 = (const int4*)(eidx + E);
    const int4* y4 = (const int4*)etype;
    for (int q = blockIdx.x * bdim + tid; q < nquad; q += gsize) {
      // prefetch ~16KB ahead in the three streams (gfx1250 global_prefetch_b8)
      __builtin_prefetch((const void*)(h4 + q + 1024), 0, 1);
      __builtin_prefetch((const void*)(t4 + q + 1024), 0, 1);
      __builtin_prefetch((const void*)(y4 + q + 1024), 0, 1);
      int4 hv = h4[q];
      int4 tv = t4[q];
      int4 yv = y4[q];
      process_edge(hv.x, tv.x, yv.x, R, shist, skey, sval, Mq);
      process_edge(hv.y, tv.y, yv.y, R, shist, skey, sval, Mq);
      process_edge(hv.z, tv.z, yv.z, R, shist, skey, sval, Mq);
      process_edge(hv.w, tv.w, yv.w, R, shist, skey, sval, Mq);
    }
  } else {
    for (int e = blockIdx.x * bdim + tid; e < E; e += gsize) {
      process_edge(eidx[e], eidx[E + e], etype[e], R, shist, skey, sval, Mq);
    }
  }
  __syncthreads();
  for (int i = tid; i < R; i += bdim)
    if (shist[i]) atomicAdd(&cnt[i], shist[i]);
}

// ---------------------------------------------------------------------------
// Kernel 4: per batch row b (one block each):
//   deg, masked-mean entity embedding (skipping zero-weight relations),
//   query-relation embedding gather, graph stats -> x[b, 0:2D+4]
// ---------------------------------------------------------------------------
__global__ void k_buildx(const float* __restrict__ emb, const int* __restrict__ qrels,
                         const int* __restrict__ qents, const int* __restrict__ pN,
                         const uint32_t* __restrict__ cnt,
                         const int* __restrict__ gkey, const int* __restrict__ gval,
                         const float* __restrict__ Mq, float* __restrict__ x,
                         int B, int R, int D, int E, int din) {
  __shared__ float wrow[TBL];
  __shared__ float part[256];
  __shared__ float sdeg;
  __shared__ int srep, sqr;

  int b = blockIdx.x, tid = threadIdx.x, bdim = blockDim.x;
  if (tid == 0) {
    sdeg = 0.f;
    int e = qents[b];
    int rep = -1;
    uint32_t idx = hash_node((uint32_t)e);
    for (;;) {
      int k = gkey[idx];
      if (k == e) { rep = gval[idx]; break; }
      if (k == -1) break;
      idx = (idx + 1) & (TBL - 1);
    }
    srep = rep;
    sqr = qrels[b];
  }
  __syncthreads();

  int rep = srep;
  float local = 0.f;
  for (int r = tid; r < R; r += bdim) {
    float w = (rep >= 0) ? Mq[(size_t)rep * R + r] : 0.f;
    wrow[r] = w;
    local += w;
  }
  atomicAdd(&sdeg, local);
  __syncthreads();

  int d = tid % D;       // D = 64
  int rs = tid / D;      // 0..3
  int nrs = bdim / D;    // 4
  float acc = 0.f;
  for (int r = rs; r < R; r += nrs) {
    float w = wrow[r];
    if (w != 0.f) acc += w * emb[((size_t)b * R + r) * D + d];
  }
  part[tid] = acc;
  __syncthreads();

  float deg = sdeg;
  if (tid < D) {
    float s = 0.f;
    for (int j = 0; j < nrs; ++j) s += part[j * D + tid];
    float ev = (deg > 0.f) ? s / fmaxf(deg, 1.f) : 0.f;
    x[(size_t)b * din + D + tid] = ev;                                // entity_emb
    x[(size_t)b * din + tid] = emb[((size_t)b * R + sqr) * D + tid];  // qrel_emb
  }
  if (tid == 0) {
    double N = (double)pN[0];
    float freq = fminf((float)cnt[sqr] / (float)E, 1.f);
    float degn = fminf(deg / (float)E, 1.f);
    double dens = (double)E / (N * N);
    float fd = (float)(dens < 1.0 ? dens : 1.0);
    float* sp = x + (size_t)b * din + 2 * D;
    sp[0] = freq;
    sp[1] = degn;
    sp[2] = freq;
    sp[3] = fd;
  }
}

// ---------------------------------------------------------------------------
// Kernel 5: fused 4-layer MLP. Layers 1-3 use V_WMMA_F32_16X16X4_F32.
// Weights staged in LDS (shared by all 16 M-tiles); activations in LDS.
//
// f32 WMMA fragment layouts (ISA 7.12.2):
//   A (16x4):  lane m=lane&15 holds A[m][kb], A[m][kb+1], kb = (lane>>4)*2
//   B (4x16):  lane n=lane&15 holds B[kb][n], B[kb+1][n]   (row-striped rule)
//   C/D (16x16): VGPR i -> row = i + 8*(lane>>4), col = lane&15
// ---------------------------------------------------------------------------
__device__ __forceinline__ void wmma_layer(const float* A, int lda,
                                           const float* Bm, int ldb,
                                           const float* bias,
                                           float* Dst, int ldd,
                                           int M, int N, int K,
                                           int wave, int nwaves, int lane) {
  int ntN = N >> 4;
  int nTiles = (M >> 4) * ntN;
  int ksteps = K >> 2;
  int mi = lane & 15;
  int half = lane >> 4;
  for (int t = wave; t < nTiles; t += nwaves) {
    int mt = t / ntN;
    int nt = t - mt * ntN;
    v8f c = {0.f, 0.f, 0.f, 0.f, 0.f, 0.f, 0.f, 0.f};
    int m = (mt << 4) + mi;
    int n = (nt << 4) + mi;
    const float* arow = A + (size_t)m * lda;
    for (int ks = 0; ks < ksteps; ++ks) {
      int k0 = (ks << 2) + (half << 1);
      v2f a;
      a.x = arow[k0];
      a.y = arow[k0 + 1];
      v2f bb;
      bb.x = Bm[k0 * ldb + n];
      bb.y = Bm[(k0 + 1) * ldb + n];
      c = __builtin_amdgcn_wmma_f32_16x16x4_f32(
          /*neg_a=*/false, a, /*neg_b=*/false, bb,
          /*c_mod=*/(short)0, c, /*reuse_a=*/false, /*reuse_b=*/false);
    }
    float bv = bias[n];
#pragma unroll
    for (int i = 0; i < 8; ++i) {
      int row = (mt << 4) + i + (half << 3);
      float v = c[i] + bv;
      Dst[(size_t)row * ldd + n] = v > 0.f ? v : 0.f;  // ReLU
    }
  }
}

__global__ void k_mlp(const float* __restrict__ x,
                      const float* __restrict__ W1, const float* __restrict__ b1,
                      const float* __restrict__ W2, const float* __restrict__ b2,
                      const float* __restrict__ W3, const float* __restrict__ b3,
                      const float* __restrict__ W4, const float* __restrict__ b4,
                      float* __restrict__ out, int B, int din, int D) {
  extern __shared__ float smem[];
  float* h1 = smem;                        // B * D
  float* h2 = h1 + (size_t)B * D;          // B * D/2
  float* h3 = h2 + (size_t)B * (D >> 1);   // B * D/4
  float* w1 = h3 + (size_t)B * (D >> 2);   // din * D
  float* w2 = w1 + (size_t)din * D;        // D * D/2
  float* w3 = w2 + (size_t)D * (D >> 1);   // D/2 * D/4

  int tid = threadIdx.x, bdim = blockDim.x;
  int lane = tid & 31;
  int wave = tid >> 5;
  int nwaves = bdim >> 5;

  // Stage all weights in LDS once (43 KB): B-fragments become ds_loads.
  int n1 = din * D, n2 = D * (D >> 1), n3 = (D >> 1) * (D >> 2);
  for (int i = tid; i < n1; i += bdim) w1[i] = W1[i];
  for (int i = tid; i < n2; i += bdim) w2[i] = W2[i];
  for (int i = tid; i < n3; i += bdim) w3[i] = W3[i];
  __syncthreads();

  // L1: [B, din] x [din, D]          (din = 2D+4 = 132, 33 k-steps)
  wmma_layer(x, din, w1, D, b1, h1, D, B, D, din, wave, nwaves, lane);
  __syncthreads();
  // L2: [B, D] x [D, D/2]
  wmma_layer(h1, D, w2, D >> 1, b2, h2, D >> 1, B, D >> 1, D, wave, nwaves, lane);
  __syncthreads();
  // L3: [B, D/2] x [D/2, D/4]
  wmma_layer(h2, D >> 1, w3, D >> 2, b3, h3, D >> 2, B, D >> 2, D >> 1, wave, nwaves, lane);
  __syncthreads();
  // L4: [B, D/4] x [D/4, 1] + sigmoid  (scalar, D/4 = 16)
  if (tid < B) {
    int Kq = D >> 2;
    float acc = b4[0];
    const float* row = h3 + (size_t)tid * Kq;
    for (int j = 0; j < Kq; ++j) acc += row[j] * W4[j];
    out[tid] = 1.f / (1.f + __expf(-acc));
  }
}

// ---------------------------------------------------------------------------
// Host launcher
// ---------------------------------------------------------------------------
extern "C" void kernel_launch(void* const* d_in, const int* in_sizes, int n_in,
                              void* d_out, int out_size, void* d_ws, size_t ws_size,
                              hipStream_t stream) {
  const float* emb   = (const float*)d_in[0];
  const int*   qrels = (const int*)d_in[1];
  const int*   qents = (const int*)d_in[2];
  const int*   eidx  = (const int*)d_in[3];
  const int*   etype = (const int*)d_in[4];
  const int*   pN    = (const int*)d_in[5];
  const float* W1 = (const float*)d_in[7];
  const float* b1 = (const float*)d_in[8];
  const float* W2 = (const float*)d_in[9];
  const float* b2 = (const float*)d_in[10];
  const float* W3 = (const float*)d_in[11];
  const float* b3 = (const float*)d_in[12];
  const float* W4 = (const float*)d_in[13];
  const float* b4 = (const float*)d_in[14];

  int B = in_sizes[1];            // 256
  int D = in_sizes[8];            // 64
  int R = in_sizes[0] / (B * D);  // 512
  int E = in_sizes[4];            // 3,200,000
  int din = in_sizes[7] / D;      // 132

  // workspace layout
  char* ws = (char*)d_ws;
  float*    Mq  = (float*)ws;                               // B*R floats
  uint32_t* cnt = (uint32_t*)(ws + (size_t)B * R * 4);      // R uints
  int*      key = (int*)((char*)cnt + (size_t)R * 4);       // TBL ints
  int*      val = key + TBL;                                // TBL ints
  float*    x   = (float*)(val + TBL);                      // B*din floats
  float*    out = (float*)d_out;

  int zwords = B * R + R;
  k_zero<<<(zwords + 255) / 256, 256, 0, stream>>>((uint32_t*)Mq, zwords);
  k_table<<<1, TBL, 0, stream>>>(qents, B, key, val);

  int eblocks = ((E + 3) / 4 + 255) / 256;
  if (eblocks > 2048) eblocks = 2048;
  k_edges<<<eblocks, 256, 0, stream>>>(eidx, etype, E, R, key, val, Mq, cnt);

  k_buildx<<<B, 256, 0, stream>>>(emb, qrels, qents, pN, cnt, key, val, Mq, x,
                                  B, R, D, E, din);

  size_t lds_floats = (size_t)B * D + (size_t)B * (D / 2) + (size_t)B * (D / 4)
                    + (size_t)din * D + (size_t)D * (D / 2)
                    + (size_t)(D / 2) * (D / 4);
  k_mlp<<<1, 256, lds_floats * sizeof(float), stream>>>(x, W1, b1, W2, b2, W3, b3,
                                                        W4, b4, out, B, din, D);
}